// pRNN_31215822307728
// MI455X (gfx1250) — compile-verified
//
#include <hip/hip_runtime.h>

#define OBS   128
#define ACT   16
#define HID   1024
#define NB    16
#define LSEQ  2048
#define INSZ  (OBS + ACT)          // 144
#define NWG   16                   // persistent workgroups in recurrence
#define CPW   (HID / NWG)          // 64 output columns per WG
#define HH_STRIDE 1028             // 1024 + 4 pad dwords -> conflict-free B frags
#define IH_STRIDE 148              // 144 + 4 pad dwords

typedef float v2f __attribute__((ext_vector_type(2)));
typedef float v8f __attribute__((ext_vector_type(8)));

static __device__ __forceinline__ v8f wmma_f32(v2f a, v2f b, v8f c) {
  // D = A(16x4) * B(4x16) + C(16x16), fp32, wave32
  return __builtin_amdgcn_wmma_f32_16x16x4_f32(false, a, false, b, (short)0, c,
                                               false, false);
}

// ---- gfx1250 async copy global->LDS (ASYNCcnt path), with fallback ---------
#if __has_builtin(__builtin_amdgcn_global_load_async_to_lds_b128)
#define HAVE_ASYNC_LDS 1
typedef int v4i __attribute__((vector_size(4 * sizeof(int))));
typedef __attribute__((address_space(1))) v4i glob_v4i;
typedef __attribute__((address_space(3))) v4i lds_v4i;
static __device__ __forceinline__ void async_cp16(const float4* g, float4* l) {
  __builtin_amdgcn_global_load_async_to_lds_b128((glob_v4i*)g, (lds_v4i*)l,
                                                 0, 0);
}
static __device__ __forceinline__ void async_wait0() {
#if __has_builtin(__builtin_amdgcn_s_wait_asynccnt)
  __builtin_amdgcn_s_wait_asynccnt(0);
#else
  asm volatile("s_wait_asynccnt 0x0" ::: "memory");
#endif
}
#else
#define HAVE_ASYNC_LDS 0
#endif

// ---------------------------------------------------------------------------
// Init: zero h ping-pong buffers and per-step arrive counters (every launch,
// so graph replay is deterministic).
// ---------------------------------------------------------------------------
__global__ void init_ws(float* __restrict__ h_bufs, int* __restrict__ counters) {
  const int stride = gridDim.x * blockDim.x;
  const int tid = blockIdx.x * blockDim.x + threadIdx.x;
  for (int i = tid; i < 2 * NB * HID; i += stride) h_bufs[i] = 0.0f;
  for (int i = tid; i < LSEQ; i += stride) counters[i] = 0;
}

// ---------------------------------------------------------------------------
// Persistent recurrence: 16 WGs x 128 threads (4 waves). Each WG owns 64
// columns of W_hh (resident in LDS, f32, padded rows) and produces its
// 16x64 slice of h each step. xproj (K=144) is fused in. Steps are separated
// by a cluster barrier (NOP when not cluster-dispatched) + a global
// arrive/wait barrier over the 16 WGs.
// ---------------------------------------------------------------------------
__global__ __launch_bounds__(128)
void rnn_recurrence(const float* __restrict__ obs,   // (NB, LSEQ+1, OBS)
                    const float* __restrict__ act,   // (NB, LSEQ, ACT)
                    const float* __restrict__ W_ih,  // (HID, INSZ)
                    const float* __restrict__ W_hh,  // (HID, HID)
                    const float* __restrict__ bias,  // (HID)
                    const float* __restrict__ noise, // (LSEQ, HID)
                    float* __restrict__ h_out,       // (NB, LSEQ, HID) = d_out h_t
                    float* __restrict__ h_bufs,      // ws: 2 * NB * HID
                    int* __restrict__ counters)      // ws: LSEQ
{
  extern __shared__ float lds[];
  float* lds_hh = lds;                       // 64 x HH_STRIDE
  float* lds_ih = lds + CPW * HH_STRIDE;     // 64 x IH_STRIDE

  const int tid   = threadIdx.x;
  const int wg    = blockIdx.x;              // 0..15
  const int jbase = wg * CPW;

  // Stage W_hh / W_ih column slices into LDS.
#if HAVE_ASYNC_LDS
  for (int r = 0; r < CPW; ++r) {
    const float4* src = (const float4*)(W_hh + (size_t)(jbase + r) * HID);
    float4* dst = (float4*)(lds_hh + r * HH_STRIDE);
    for (int c = tid; c < HID / 4; c += 128) async_cp16(src + c, dst + c);
  }
  for (int r = 0; r < CPW; ++r) {
    const float4* src = (const float4*)(W_ih + (size_t)(jbase + r) * INSZ);
    float4* dst = (float4*)(lds_ih + r * IH_STRIDE);
    for (int c = tid; c < INSZ / 4; c += 128) async_cp16(src + c, dst + c);
  }
  async_wait0();
#else
  for (int r = 0; r < CPW; ++r) {
    const float4* src = (const float4*)(W_hh + (size_t)(jbase + r) * HID);
    float4* dst = (float4*)(lds_hh + r * HH_STRIDE);
    for (int c = tid; c < HID / 4; c += 128) dst[c] = src[c];
  }
  for (int r = 0; r < CPW; ++r) {
    const float4* src = (const float4*)(W_ih + (size_t)(jbase + r) * INSZ);
    float4* dst = (float4*)(lds_ih + r * IH_STRIDE);
    for (int c = tid; c < INSZ / 4; c += 128) dst[c] = src[c];
  }
#endif
  __syncthreads();

  const int lane  = tid & 31;
  const int wave  = tid >> 5;                // 0..3 -> 4 col tiles of 16
  const int half  = lane >> 4;               // K-half selector for A/B frags
  const int lr    = lane & 15;               // M (A row / batch) or N (B col)
  const int jc    = wave * 16 + lr;          // local output column
  const int jglob = jbase + jc;

  const float bj = bias[jglob];
  const float* __restrict__ wrow  = lds_hh + jc * HH_STRIDE;  // B: W_hh[jglob][*]
  const float* __restrict__ wirow = lds_ih + jc * IH_STRIDE;  // B: W_ih[jglob][*]
  const float* __restrict__ obs_n = obs + (size_t)lr * ((LSEQ + 1) * OBS);
  const float* __restrict__ act_n = act + (size_t)lr * (LSEQ * ACT);

  for (int l = 0; l < LSEQ; ++l) {
    const float* __restrict__ hprev = h_bufs + (l & 1) * (NB * HID);
    float* __restrict__       hnext = h_bufs + ((l + 1) & 1) * (NB * HID);

    v8f acc0 = {}, acc1 = {}, acc2 = {}, acc3 = {};

    // ---- fused xproj: x_l(16x144) @ W_ih_slice^T -> 16x16 tile ----
    const float* __restrict__ obs_l = obs_n + (size_t)l * OBS;
    const float* __restrict__ act_l = act_n + (size_t)l * ACT;
    #pragma unroll 3
    for (int k0 = 0; k0 < INSZ; k0 += 8) {
      const int k = k0 + 2 * half;
      v2f a0, b0, a1, b1;
      const float* p0 = (k     < OBS) ? (obs_l + k)     : (act_l + (k - OBS));
      const float* p1 = (k + 1 < OBS) ? (obs_l + k + 1) : (act_l + (k + 1 - OBS));
      const float* p2 = (k + 4 < OBS) ? (obs_l + k + 4) : (act_l + (k + 4 - OBS));
      const float* p3 = (k + 5 < OBS) ? (obs_l + k + 5) : (act_l + (k + 5 - OBS));
      a0.x = *p0; a0.y = *p1;
      a1.x = *p2; a1.y = *p3;
      b0.x = wirow[k];     b0.y = wirow[k + 1];
      b1.x = wirow[k + 4]; b1.y = wirow[k + 5];
      acc0 = wmma_f32(a0, b0, acc0);
      acc1 = wmma_f32(a1, b1, acc1);
    }

    // ---- recurrent GEMM: h_prev(16x1024) @ W_hh_slice^T, 4 accumulators ----
    const float* __restrict__ hrow = hprev + lr * HID;  // A row m = lr
    #pragma unroll 4
    for (int k0 = 0; k0 < HID; k0 += 16) {
      const int k = k0 + 2 * half;
      v2f a0, b0, a1, b1, a2, b2, a3, b3;
      a0.x = hrow[k];      a0.y = hrow[k + 1];
      b0.x = wrow[k];      b0.y = wrow[k + 1];
      a1.x = hrow[k + 4];  a1.y = hrow[k + 5];
      b1.x = wrow[k + 4];  b1.y = wrow[k + 5];
      a2.x = hrow[k + 8];  a2.y = hrow[k + 9];
      b2.x = wrow[k + 8];  b2.y = wrow[k + 9];
      a3.x = hrow[k + 12]; a3.y = hrow[k + 13];
      b3.x = wrow[k + 12]; b3.y = wrow[k + 13];
      acc0 = wmma_f32(a0, b0, acc0);
      acc1 = wmma_f32(a1, b1, acc1);
      acc2 = wmma_f32(a2, b2, acc2);
      acc3 = wmma_f32(a3, b3, acc3);
    }

    // bias + noise (column-only) + relu, write h slice (coalesced per VGPR)
    const float s = bj + noise[(size_t)l * HID + jglob];
    #pragma unroll
    for (int r = 0; r < 8; ++r) {
      float v = (acc0[r] + acc1[r]) + (acc2[r] + acc3[r]) + s;
      v = v > 0.0f ? v : 0.0f;
      const int m = r + 8 * half;                 // batch row
      hnext[m * HID + jglob] = v;
      h_out[((size_t)m * LSEQ + l) * HID + jglob] = v;
    }

    // Prefetch next step's activations while we sit in the barrier.
    if (l + 1 < LSEQ) {
      __builtin_prefetch(obs_n + (size_t)(l + 1) * OBS, 0, 3);
      __builtin_prefetch(act_n + (size_t)(l + 1) * ACT, 0, 3);
      __builtin_prefetch(noise + (size_t)(l + 1) * HID + jglob, 0, 3);
    }

    // ---- step barrier across the 16 persistent WGs ----
    __threadfence();          // release h slice to device scope
    __syncthreads();
#if __has_builtin(__builtin_amdgcn_s_cluster_barrier)
    // NOP unless this 16-WG grid is dispatched as a cluster (ISA 3.2); the
    // atomic counter below carries correctness either way.
    __builtin_amdgcn_s_cluster_barrier();
#endif
    if (tid == 0) {
      __hip_atomic_fetch_add(&counters[l], 1, __ATOMIC_RELEASE,
                             __HIP_MEMORY_SCOPE_AGENT);
      while (__hip_atomic_load(&counters[l], __ATOMIC_ACQUIRE,
                               __HIP_MEMORY_SCOPE_AGENT) < NWG)
        __builtin_amdgcn_s_sleep(1);
    }
    __syncthreads();
    __threadfence();          // acquire other WGs' h slices
  }
}

// ---------------------------------------------------------------------------
// Output head: y = sigmoid(h_t @ W_out^T). One 16-row tile of (n*L+l) per
// block; 8 waves cover the 8 OBS column tiles, sharing A rows. K = 1024.
// ---------------------------------------------------------------------------
__global__ __launch_bounds__(256)
void rnn_output(const float* __restrict__ h_t,    // (NB*LSEQ, HID) row-major
                const float* __restrict__ W_out,  // (OBS, HID)
                float* __restrict__ y)            // (NB*LSEQ, OBS)
{
  const int tid  = threadIdx.x;
  const int lane = tid & 31;
  const int wave = tid >> 5;                // 0..7 -> col tile
  const int half = lane >> 4;
  const int lr   = lane & 15;
  const size_t mbase = (size_t)blockIdx.x * 16;
  const int ot = wave * 16;

  const float* __restrict__ arow = h_t + (mbase + lr) * HID;
  const float* __restrict__ brow = W_out + (size_t)(ot + lr) * HID;

  v8f acc0 = {}, acc1 = {}, acc2 = {}, acc3 = {};
  #pragma unroll 4
  for (int k0 = 0; k0 < HID; k0 += 16) {
    const int k = k0 + 2 * half;
    v2f a0, b0, a1, b1, a2, b2, a3, b3;
    a0.x = arow[k];      a0.y = arow[k + 1];
    b0.x = brow[k];      b0.y = brow[k + 1];
    a1.x = arow[k + 4];  a1.y = arow[k + 5];
    b1.x = brow[k + 4];  b1.y = brow[k + 5];
    a2.x = arow[k + 8];  a2.y = arow[k + 9];
    b2.x = brow[k + 8];  b2.y = brow[k + 9];
    a3.x = arow[k + 12]; a3.y = arow[k + 13];
    b3.x = brow[k + 12]; b3.y = brow[k + 13];
    acc0 = wmma_f32(a0, b0, acc0);
    acc1 = wmma_f32(a1, b1, acc1);
    acc2 = wmma_f32(a2, b2, acc2);
    acc3 = wmma_f32(a3, b3, acc3);
  }

  #pragma unroll
  for (int r = 0; r < 8; ++r) {
    const float v = (acc0[r] + acc1[r]) + (acc2[r] + acc3[r]);
    const float sg = 1.0f / (1.0f + __expf(-v));
    y[(mbase + r + 8 * half) * OBS + ot + lr] = sg;
  }
}

// ---------------------------------------------------------------------------
// obs_target = obs[:, 1:, :]  (per-n contiguous block copy, float4)
// ---------------------------------------------------------------------------
__global__ void copy_obs_target(const float4* __restrict__ obs4,
                                float4* __restrict__ tgt4) {
  const size_t per_n  = (size_t)LSEQ * OBS / 4;          // 65536
  const size_t src_st = (size_t)(LSEQ + 1) * OBS / 4;    // 65568
  const size_t total  = (size_t)NB * per_n;
  const size_t stride = (size_t)gridDim.x * blockDim.x;
  for (size_t i = (size_t)blockIdx.x * blockDim.x + threadIdx.x; i < total;
       i += stride) {
    const size_t n = i / per_n;
    const size_t j = i - n * per_n;
    tgt4[i] = obs4[n * src_st + (OBS / 4) + j];
  }
}

// ---------------------------------------------------------------------------
extern "C" void kernel_launch(void* const* d_in, const int* in_sizes, int n_in,
                              void* d_out, int out_size, void* d_ws,
                              size_t ws_size, hipStream_t stream) {
  const float* obs   = (const float*)d_in[0];
  const float* act   = (const float*)d_in[1];
  const float* W_ih  = (const float*)d_in[2];
  const float* W_hh  = (const float*)d_in[3];
  const float* bias  = (const float*)d_in[4];
  const float* W_out = (const float*)d_in[5];
  const float* noise = (const float*)d_in[6];

  float* y   = (float*)d_out;                       // (NB, LSEQ, OBS)
  float* h_t = y + (size_t)NB * LSEQ * OBS;         // (NB, LSEQ, HID)
  float* tgt = h_t + (size_t)NB * LSEQ * HID;       // (NB, LSEQ, OBS)

  float* h_bufs  = (float*)d_ws;                    // 2*NB*HID floats (128 KB)
  int*   counters = (int*)((char*)d_ws + (size_t)2 * NB * HID * sizeof(float));

  init_ws<<<64, 256, 0, stream>>>(h_bufs, counters);

  const size_t lds_bytes =
      (size_t)(CPW * HH_STRIDE + CPW * IH_STRIDE) * sizeof(float);  // 301056 B
  rnn_recurrence<<<NWG, 128, lds_bytes, stream>>>(
      obs, act, W_ih, W_hh, bias, noise, h_t, h_bufs, counters);

  rnn_output<<<(NB * LSEQ) / 16, 256, 0, stream>>>(h_t, W_out, y);

  copy_obs_target<<<1024, 256, 0, stream>>>((const float4*)obs, (float4*)tgt);
}